// PersonalizedPageRankGraphAttentionLayer_20968030339149
// MI455X (gfx1250) — compile-verified
//
#include <hip/hip_runtime.h>
#include <hip/hip_bf16.h>
#include <math.h>

#define Nn    2048
#define Nn4   (Nn / 4)
#define FIN   512
#define FOUT  256
#define TOPK  32
#define CAP   96        // max in/out-degree we store (mean ~17, binomial tail << CAP)
#define MAXIT 200
#define ALPHA 0.25f
#define EPSV  1e-4f

typedef __attribute__((ext_vector_type(2))) float v2f;
typedef __attribute__((ext_vector_type(8))) float v8f;

// ---------------------------------------------------------------------------
// Init: RT = alpha * I (transposed residual, RT[t,s]=R[s,t]), PT = 0.
// float4 lanes: diagonal element lands in component (s & 3).
// ---------------------------------------------------------------------------
__global__ void init_pr(float4* __restrict__ RT, float4* __restrict__ PT) {
    unsigned i = blockIdx.x * 256u + threadIdx.x;      // i < Nn*Nn/4
    unsigned t = i / Nn4, s4 = i % Nn4;
    float4 z = make_float4(0.f, 0.f, 0.f, 0.f);
    float4 r = z;
    if (t >> 2 == s4 >> 0 * 0 + 0 && false) {}          // (no-op; clarity below)
    if ((t >> 2) == (s4 >> 0) && false) {}
    // diagonal: RT[t, s] = alpha when s == t  ->  s4 == t/4, comp == t&3
    if (s4 == (t >> 2)) {
        ((float*)&r)[t & 3] = ALPHA;
    }
    RT[i] = r;
    PT[i] = z;
}

// ---------------------------------------------------------------------------
// Per-row degree + out-neighbor (row) CSR. Block per row, coalesced scan.
// ---------------------------------------------------------------------------
__global__ void row_deg_csr(const float* __restrict__ adj, float* __restrict__ deg,
                            int* __restrict__ rowCnt, int* __restrict__ rowIdx) {
    int r = blockIdx.x;
    __shared__ float sred[256];
    __shared__ int scnt;
    if (threadIdx.x == 0) scnt = 0;
    __syncthreads();
    float sum = 0.0f;
    for (int c = threadIdx.x; c < Nn; c += 256) {
        float v = adj[(size_t)r * Nn + c];
        sum += v;
        if (v > 0.0f) {
            int p = atomicAdd(&scnt, 1);
            if (p < CAP) rowIdx[r * CAP + p] = c;
        }
    }
    sred[threadIdx.x] = sum;
    __syncthreads();
    for (int o = 128; o > 0; o >>= 1) {
        if (threadIdx.x < o) sred[threadIdx.x] += sred[threadIdx.x + o];
        __syncthreads();
    }
    if (threadIdx.x == 0) { deg[r] = sred[0]; rowCnt[r] = min(scnt, CAP); }
}

// ---------------------------------------------------------------------------
// In-neighbor (column) CSR: thread owns column v, streams rows (coalesced).
// ---------------------------------------------------------------------------
__global__ void col_csr(const float* __restrict__ adj,
                        int* __restrict__ colCnt, int* __restrict__ colIdx) {
    int v = blockIdx.x * 256 + threadIdx.x;
    if (v >= Nn) return;
    int c = 0;
    for (int u = 0; u < Nn; ++u) {
        if (adj[(size_t)u * Nn + v] > 0.0f) {
            if (c < CAP) colIdx[v * CAP + c] = u;
            c++;
        }
    }
    colCnt[v] = min(c, CAP);
}

// thr = alpha*eps*deg ; qsc = (1-alpha)/deg ; dgs = sqrt(deg) ; flags reset
__global__ void scalars_k(const float* __restrict__ deg, float* __restrict__ thr,
                          float* __restrict__ qsc, float* __restrict__ dgs,
                          int* __restrict__ flags) {
    int i = blockIdx.x * 256 + threadIdx.x;
    if (i < Nn) {
        float d = deg[i];
        thr[i] = ALPHA * EPSV * d;
        qsc[i] = (1.0f - ALPHA) / d;
        dgs[i] = sqrtf(fmaxf(d, 1e-12f));
    }
    if (i == 0) { flags[0] = 0; flags[1] = 0; }
}

// ---------------------------------------------------------------------------
// PPR kernel 1 (float4 / b128): push detection.
//   QT[u,s] = (RT>=thr[u]) ? RT * (1-a)/deg[u] : 0 ;  PT += push ; count pushes
// ---------------------------------------------------------------------------
__global__ void push_k(const float4* __restrict__ RT, float4* __restrict__ PT,
                       float4* __restrict__ QT, const float* __restrict__ thr,
                       const float* __restrict__ qsc, int* __restrict__ flags) {
    if (flags[1]) return;                       // converged: no-op
    int u = blockIdx.x;
    int s4 = blockIdx.y * 256 + threadIdx.x;    // gridDim.y = Nn4/256 = 2
    __shared__ int s_any;
    if (threadIdx.x == 0) s_any = 0;
    __syncthreads();
    size_t o = (size_t)u * Nn4 + s4;
    float4 r = RT[o];
    float T = thr[u], q = qsc[u];
    bool px = (r.x >= T), py = (r.y >= T), pz = (r.z >= T), pw = (r.w >= T);
    float4 qv;
    qv.x = px ? r.x * q : 0.0f;
    qv.y = py ? r.y * q : 0.0f;
    qv.z = pz ? r.z * q : 0.0f;
    qv.w = pw ? r.w * q : 0.0f;
    QT[o] = qv;
    if (px | py | pz | pw) {
        float4 p4 = PT[o];
        if (px) p4.x += r.x;
        if (py) p4.y += r.y;
        if (pz) p4.z += r.z;
        if (pw) p4.w += r.w;
        PT[o] = p4;
        s_any = 1;
    }
    __syncthreads();
    if (threadIdx.x == 0 && s_any) atomicAdd(&flags[0], 1);
}

__global__ void check_k(int* __restrict__ flags) {
    if (flags[0] == 0) flags[1] = 1;            // nothing above threshold -> done
    flags[0] = 0;
}

// ---------------------------------------------------------------------------
// PPR kernel 2 (float4 / b128): sparse spread (L2-resident gather SpMM).
//   RT[v,s] = (RT>=thr[v] ? 0 : RT) + sum_{u in in(v)} QT[u,s]
// ---------------------------------------------------------------------------
__global__ void spread_k(float4* __restrict__ RT, const float4* __restrict__ QT,
                         const int* __restrict__ colCnt, const int* __restrict__ colIdx,
                         const float* __restrict__ thr, const int* __restrict__ flags) {
    if (flags[1]) return;
    int v = blockIdx.x;
    int s4 = blockIdx.y * 256 + threadIdx.x;
    int cnt = colCnt[v];
    float4 acc = make_float4(0.f, 0.f, 0.f, 0.f);
    for (int j = 0; j < cnt; ++j) {
        int u = colIdx[v * CAP + j];
        float4 qv = QT[(size_t)u * Nn4 + s4];   // lane-contiguous b128, L2-resident
        acc.x += qv.x; acc.y += qv.y; acc.z += qv.z; acc.w += qv.w;
    }
    size_t o = (size_t)v * Nn4 + s4;
    float4 old = RT[o];
    float T = thr[v];
    float4 res;
    res.x = (old.x >= T ? 0.0f : old.x) + acc.x;
    res.y = (old.y >= T ? 0.0f : old.y) + acc.y;
    res.z = (old.z >= T ? 0.0f : old.z) + acc.z;
    res.w = (old.w >= T ? 0.0f : old.w) + acc.w;
    RT[o] = res;
}

// ---------------------------------------------------------------------------
// HW = h @ W via v_wmma_f32_16x16x4_f32. One wave per 16x16 tile, K-loop of 4.
// A 16x4 layout: lanes 0-15 rows M, v[0]=K0/K2, v[1]=K1/K3 (hi lanes K2/K3).
// B 4x16 layout: lanes hold column N, v[0]=K0/K2, v[1]=K1/K3.
// C/D 16x16:     lane N = lane&15, VGPR i -> M = (lane>>4)*8 + i.
// ---------------------------------------------------------------------------
__global__ void hw_gemm(const float* __restrict__ h, const float* __restrict__ W,
                        float* __restrict__ HW) {
    int nt  = blockIdx.x;                 // FOUT/16 tiles
    int mt  = blockIdx.y;                 // Nn/16 tiles
    int lane = threadIdx.x;               // 32 = one wave
    int grp = lane >> 4;
    int lr  = lane & 15;
    int m = mt * 16 + lr;
    int n = nt * 16 + lr;
    v8f c = {};
    for (int k = 0; k < FIN; k += 4) {
        int ka = k + grp * 2;
        v2f a, b;
        a.x = h[(size_t)m * FIN + ka];
        a.y = h[(size_t)m * FIN + ka + 1];
        b.x = W[(size_t)ka * FOUT + n];
        b.y = W[(size_t)(ka + 1) * FOUT + n];
        c = __builtin_amdgcn_wmma_f32_16x16x4_f32(false, a, false, b,
                                                  (short)0, c, false, false);
    }
    for (int i = 0; i < 8; ++i)
        HW[(size_t)(mt * 16 + grp * 8 + i) * FOUT + nt * 16 + lr] = c[i];
}

// h1[s] = HW[s,:] . a[:FOUT]  ;  h2[s] = HW[s,:] . a[FOUT:]
__global__ void h12_k(const float* __restrict__ HW, const float* __restrict__ a,
                      float* __restrict__ h1, float* __restrict__ h2) {
    int s = blockIdx.x, f = threadIdx.x;  // 256 threads
    __shared__ float r1[256], r2[256];
    float p = HW[(size_t)s * FOUT + f];
    r1[f] = p * a[f];
    r2[f] = p * a[FOUT + f];
    __syncthreads();
    for (int o = 128; o > 0; o >>= 1) {
        if (f < o) { r1[f] += r1[f + o]; r2[f] += r2[f + o]; }
        __syncthreads();
    }
    if (f == 0) { h1[s] = r1[0]; h2[s] = r2[0]; }
}

// ---------------------------------------------------------------------------
// Per-source top-32 of P (scanning PT columns = lane-coalesced) + symmetric
// degree rescale: val * sqrt(deg[s]) / sqrt(deg[t]).
// ---------------------------------------------------------------------------
__global__ void topk_k(const float* __restrict__ PT, const float* __restrict__ dgs,
                       float* __restrict__ tkV, int* __restrict__ tkI) {
    int s = blockIdx.x * 256 + threadIdx.x;
    if (s >= Nn) return;
    float vals[TOPK]; int idxs[TOPK];
    for (int j = 0; j < TOPK; ++j) { vals[j] = PT[(size_t)j * Nn + s]; idxs[j] = j; }
    float mn = vals[0]; int mj = 0;
    for (int j = 1; j < TOPK; ++j) if (vals[j] < mn) { mn = vals[j]; mj = j; }
    for (int t = TOPK; t < Nn; ++t) {
        float v = PT[(size_t)t * Nn + s];
        if (v > mn) {
            vals[mj] = v; idxs[mj] = t;
            mn = vals[0]; mj = 0;
            for (int j = 1; j < TOPK; ++j) if (vals[j] < mn) { mn = vals[j]; mj = j; }
        }
    }
    float ds = dgs[s];
    for (int j = 0; j < TOPK; ++j) {
        tkV[s * TOPK + j] = vals[j] * ds / dgs[idxs[j]];
        tkI[s * TOPK + j] = idxs[j];
    }
}

// ---------------------------------------------------------------------------
// Attention + output. Masked softmax support == adjacency neighbors only
// (exp(-9e15 - max) underflows to exact 0 in f32). One block per source row.
// ---------------------------------------------------------------------------
__global__ void attn_k(const float* __restrict__ HW, const float* __restrict__ h1,
                       const float* __restrict__ h2, const float* __restrict__ appr,
                       const float* __restrict__ tkV, const int* __restrict__ tkI,
                       const int* __restrict__ rowCnt, const int* __restrict__ rowIdx,
                       float* __restrict__ out) {
    int s = blockIdx.x, tid = threadIdx.x;     // FOUT threads
    __shared__ float s_att[CAP];
    __shared__ int   s_nbr[CAP];
    __shared__ float s_m, s_sum;
    int cnt = rowCnt[s];
    for (int j = tid; j < cnt; j += FOUT) s_nbr[j] = rowIdx[s * CAP + j];
    __syncthreads();
    if (tid < cnt) {
        int t = s_nbr[tid];
        float pv = 0.0f;
        for (int j = 0; j < TOPK; ++j)
            if (tkI[s * TOPK + j] == t) pv = tkV[s * TOPK + j];
        float e = h1[s] + h2[t] + appr[0] * pv;
        s_att[tid] = (e >= 0.0f) ? e : 0.2f * e;        // leaky_relu(0.2)
    }
    __syncthreads();
    if (tid == 0) {
        float m = -3.4e38f;
        for (int j = 0; j < cnt; ++j) m = fmaxf(m, s_att[j]);
        s_m = m;
    }
    __syncthreads();
    if (tid < cnt) s_att[tid] = expf(s_att[tid] - s_m);
    __syncthreads();
    if (tid == 0) {
        float sm = 0.0f;
        for (int j = 0; j < cnt; ++j) sm += s_att[j];
        s_sum = sm;
    }
    __syncthreads();
    float inv = 1.0f / s_sum;
    float acc = 0.0f;
    for (int j = 0; j < cnt; ++j)
        acc += s_att[j] * HW[(size_t)s_nbr[j] * FOUT + tid];
    out[(size_t)s * FOUT + tid] = acc * inv;
}

// ---------------------------------------------------------------------------
extern "C" void kernel_launch(void* const* d_in, const int* in_sizes, int n_in,
                              void* d_out, int out_size, void* d_ws, size_t ws_size,
                              hipStream_t stream) {
    const float* h    = (const float*)d_in[0];   // [N, FIN]
    const float* adj  = (const float*)d_in[1];   // [N, N]
    const float* W    = (const float*)d_in[2];   // [FIN, FOUT]
    const float* a    = (const float*)d_in[3];   // [2*FOUT]
    const float* appr = (const float*)d_in[4];   // [1]
    float* out = (float*)d_out;

    // --- workspace carve (~51 MB, fits MI455X L2 for the hot loop) ---
    const size_t NN = (size_t)Nn * Nn;
    float* RT  = (float*)d_ws;
    float* PT  = RT + NN;
    float* QT  = PT + NN;
    float* HW  = QT + NN;                 // N*FOUT
    float* deg = HW + (size_t)Nn * FOUT;
    float* thr = deg + Nn;
    float* qsc = thr + Nn;
    float* dgs = qsc + Nn;
    float* h1v = dgs + Nn;
    float* h2v = h1v + Nn;
    float* tkV = h2v + Nn;                // N*TOPK
    int* tkI    = (int*)(tkV + (size_t)Nn * TOPK);
    int* rowCnt = tkI + (size_t)Nn * TOPK;
    int* rowIdx = rowCnt + Nn;            // N*CAP
    int* colCnt = rowIdx + (size_t)Nn * CAP;
    int* colIdx = colCnt + Nn;            // N*CAP
    int* flags  = colIdx + (size_t)Nn * CAP;   // [0]=push count, [1]=converged

    // --- setup ---
    init_pr<<<NN / 4 / 256, 256, 0, stream>>>((float4*)RT, (float4*)PT);
    row_deg_csr<<<Nn, 256, 0, stream>>>(adj, deg, rowCnt, rowIdx);
    col_csr<<<Nn / 256, 256, 0, stream>>>(adj, colCnt, colIdx);
    scalars_k<<<(Nn + 255) / 256, 256, 0, stream>>>(deg, thr, qsc, dgs, flags);

    // --- fixed-shape PPR loop (device-side convergence early-out) ---
    dim3 gridNN(Nn, Nn4 / 256);           // 2048 x 2 blocks, b128 per lane
    for (int it = 0; it < MAXIT; ++it) {
        push_k<<<gridNN, 256, 0, stream>>>((const float4*)RT, (float4*)PT,
                                           (float4*)QT, thr, qsc, flags);
        check_k<<<1, 1, 0, stream>>>(flags);
        spread_k<<<gridNN, 256, 0, stream>>>((float4*)RT, (const float4*)QT,
                                             colCnt, colIdx, thr, flags);
    }

    // --- GAT head ---
    hw_gemm<<<dim3(FOUT / 16, Nn / 16), 32, 0, stream>>>(h, W, HW);
    h12_k<<<Nn, 256, 0, stream>>>(HW, a, h1v, h2v);
    topk_k<<<Nn / 256, 256, 0, stream>>>(PT, dgs, tkV, tkI);
    attn_k<<<Nn, FOUT, 0, stream>>>(HW, h1v, h2v, appr, tkV, tkI, rowCnt, rowIdx, out);
}